// myDC_14396730376409
// MI455X (gfx1250) — compile-verified
//
#include <hip/hip_runtime.h>
#include <math.h>

// ----------------------------------------------------------------------------
// MRI SENSE CG-reconstruction for MI455X (gfx1250, wave32).
//
// Strategy (see analysis): the per-sample working set (~165 MB) fits in the
// 192 MB L2, so the solve is launch/latency bound. The 3-D FFTs are expressed
// as dense DFT matmuls mapped onto V_WMMA_F32_16X16X4_F32 (fp32 matrix path),
// 4 real WMMAs per complex tile-step, data staged through LDS.
// ----------------------------------------------------------------------------

#define NX 128
#define NY 128
#define NZ 64
#define NV (NX * NY * NZ)   // 1<<20 voxels per sample
#define NCOIL 8
#define NB 2
#define CG_ITERS 10
#define CG_TOL 1e-5f

typedef float v2f __attribute__((ext_vector_type(2)));
typedef float v8f __attribute__((ext_vector_type(8)));

// -------------------------- reduction helper --------------------------------
__device__ __forceinline__ float block_reduce_add(float v) {
    __shared__ float sm[8];
    const int lane = threadIdx.x & 31;
    const int w    = threadIdx.x >> 5;
#pragma unroll
    for (int o = 16; o > 0; o >>= 1) v += __shfl_down(v, o, 32);
    if (lane == 0) sm[w] = v;
    __syncthreads();
    if (w == 0) {
        v = (lane < (int)(blockDim.x >> 5)) ? sm[lane] : 0.0f;
#pragma unroll
        for (int o = 16; o > 0; o >>= 1) v += __shfl_down(v, o, 32);
    }
    return v;   // valid in thread 0
}

// ------------------------------ twiddles ------------------------------------
// W[j*n+k] = exp(sign * 2*pi*i * j*k / n), planar re/im.
__global__ void twiddle_init(float* __restrict__ wr, float* __restrict__ wi,
                             int n, float sign) {
    const float twopi = 6.283185307179586f;
    for (int idx = blockIdx.x * blockDim.x + threadIdx.x; idx < n * n;
         idx += gridDim.x * blockDim.x) {
        int j = idx / n, k = idx - j * n;
        float ang = sign * twopi * (float)((j * k) % n) / (float)n;
        wr[idx] = __cosf(ang);
        wi[idx] = __sinf(ang);
    }
}

// ------------------------- WMMA-based batched DFT ---------------------------
// One workgroup = one (coil, 16-line tile). OUT[n,16] = W[n,n] x IN[n,16],
// complex via 4 real V_WMMA_F32_16X16X4_F32 accumulators. In-place on the
// coil volume (tile fully staged in LDS before any global store).
//
// Column address:  addr = row*rs + (cg>>cshift)*cs_hi + (cg&cmask)*cs_lo
//   axis X: rs=8192 cs_lo=1  cmask=~0 cshift=31 cs_hi=0      (cg = y*64+z)
//   axis Y: rs=64   cs_lo=1  cmask=63 cshift=6  cs_hi=8192   (cg = x*64+z)
//   axis Z: rs=1    cs_lo=64 cmask=~0 cshift=31 cs_hi=0      (cg = x*128+y)
__global__ __launch_bounds__(256)
void dft_axis_kernel(float* __restrict__ dr, float* __restrict__ di,
                     const float* __restrict__ Wr, const float* __restrict__ Wi,
                     int n, int rs, int cs_lo, int cmask, int cshift, int cs_hi,
                     int tiles_per_coil, long coil_stride) {
    __shared__ float Xr[128 * 17];   // 17-float row pitch: conflict-free
    __shared__ float Xi[128 * 17];

    const int coil = blockIdx.x / tiles_per_coil;
    const int tile = blockIdx.x - coil * tiles_per_coil;
    const long base = (long)coil * coil_stride;
    const int col0 = tile * 16;
    const int tid  = threadIdx.x;

    // Stage the full n x 16 complex tile into LDS.
    for (int idx = tid; idx < n * 16; idx += 256) {
        int row = idx >> 4, col = idx & 15;
        int cg = col0 + col;
        long ca = (long)(cg >> cshift) * cs_hi + (long)(cg & cmask) * cs_lo;
        long g = base + (long)row * rs + ca;
        Xr[row * 17 + col] = dr[g];
        Xi[row * 17 + col] = di[g];
    }
    __syncthreads();

    const int w    = tid >> 5;      // wave id (8 wave32s per block)
    const int lane = tid & 31;
    const int l    = lane & 15;
    const int h    = lane >> 4;     // half-wave select (K/M split per ISA layout)

    if (w < (n >> 4)) {             // n=64 -> waves 4..7 idle after staging
        const int arow = (w << 4) + l;          // A-matrix row (M = lane%16)
        const float* wrp = Wr + (long)arow * n;
        const float* wip = Wi + (long)arow * n;
        __builtin_prefetch(wrp, 0, 3);          // global_prefetch of twiddle row
        __builtin_prefetch(wip, 0, 3);

        v8f cRR = {}, cII = {}, cRI = {}, cIR = {};
#pragma unroll 4
        for (int k = 0; k < n; k += 4) {
            const int kk = k + (h << 1);        // lanes 16..31 carry K+2,K+3
            v2f aR, aI, bR, bI;
            aR[0] = wrp[kk]; aR[1] = wrp[kk + 1];
            aI[0] = wip[kk]; aI[1] = wip[kk + 1];
            bR[0] = Xr[kk * 17 + l]; bR[1] = Xr[(kk + 1) * 17 + l];
            bI[0] = Xi[kk * 17 + l]; bI[1] = Xi[(kk + 1) * 17 + l];
            cRR = __builtin_amdgcn_wmma_f32_16x16x4_f32(false, aR, false, bR,
                                                        (short)0, cRR, false, false);
            cII = __builtin_amdgcn_wmma_f32_16x16x4_f32(false, aI, false, bI,
                                                        (short)0, cII, false, false);
            cRI = __builtin_amdgcn_wmma_f32_16x16x4_f32(false, aR, false, bI,
                                                        (short)0, cRI, false, false);
            cIR = __builtin_amdgcn_wmma_f32_16x16x4_f32(false, aI, false, bR,
                                                        (short)0, cIR, false, false);
        }

        // C/D layout: lane%16 = N, VGPR r = M (lanes>=16: M+8).
        const int cg = col0 + l;
        const long ca = (long)(cg >> cshift) * cs_hi + (long)(cg & cmask) * cs_lo;
#pragma unroll
        for (int r = 0; r < 8; r++) {
            const int row = (w << 4) + r + (h << 3);
            const long g = base + (long)row * rs + ca;
            dr[g] = cRR[r] - cII[r];    // Re = Wr·Xr - Wi·Xi (f32 WMMA: C-neg only,
            di[g] = cRI[r] + cIR[r];    //  so combine with VALU)
        }
    }
}

// ------------------------------ CG pointwise --------------------------------
__global__ void init_vecs(const float* __restrict__ rhs,
                          float* __restrict__ xr, float* __restrict__ xi,
                          float* __restrict__ rr, float* __restrict__ ri,
                          float* __restrict__ pr, float* __restrict__ pi,
                          float* __restrict__ s) {
    float acc = 0.0f;
    for (int v = blockIdx.x * blockDim.x + threadIdx.x; v < NV;
         v += gridDim.x * blockDim.x) {
        float re = rhs[2 * v], im = rhs[2 * v + 1];
        xr[v] = 0.0f; xi[v] = 0.0f;
        rr[v] = re;  ri[v] = im;
        pr[v] = re;  pi[v] = im;
        acc += re * re + im * im;
    }
    float t = block_reduce_add(acc);
    if (threadIdx.x == 0) atomicAdd(&s[0], t);        // rTr0
}

__global__ void coil_expand(const float* __restrict__ cr, const float* __restrict__ ci,
                            const float* __restrict__ pr, const float* __restrict__ pi,
                            float* __restrict__ qr, float* __restrict__ qi) {
    for (long idx = blockIdx.x * (long)blockDim.x + threadIdx.x;
         idx < (long)NCOIL * NV; idx += (long)gridDim.x * blockDim.x) {
        long v = idx & (NV - 1);
        float a = cr[idx], b = ci[idx], x = pr[v], y = pi[v];
        qr[idx] = a * x - b * y;
        qi[idx] = a * y + b * x;
    }
}

__global__ void mask_scale(float* __restrict__ qr, float* __restrict__ qi,
                           const float* __restrict__ mask) {
    const float invN = 1.0f / (float)NV;              // ifftn normalization
    for (long idx = blockIdx.x * (long)blockDim.x + threadIdx.x;
         idx < (long)NCOIL * NV; idx += (long)gridDim.x * blockDim.x) {
        float m = mask[idx & (NV - 1)] * invN;
        qr[idx] *= m;
        qi[idx] *= m;
    }
}

// Ap = sum_c conj(csm)*q + (lam1+lam2)*p ; fused pAp reduction.
__global__ void coil_combine(const float* __restrict__ cr, const float* __restrict__ ci,
                             const float* __restrict__ qr, const float* __restrict__ qi,
                             const float* __restrict__ pr, const float* __restrict__ pi,
                             float* __restrict__ apr, float* __restrict__ api,
                             const float* __restrict__ lam1,
                             const float* __restrict__ lam2,
                             float* __restrict__ s) {
    const float lam = lam1[0] + lam2[0];
    float acc = 0.0f;
    for (int v = blockIdx.x * blockDim.x + threadIdx.x; v < NV;
         v += gridDim.x * blockDim.x) {
        float sr = 0.0f, si = 0.0f;
#pragma unroll
        for (int c = 0; c < NCOIL; c++) {
            long k = (long)c * NV + v;
            float a = cr[k], b = ci[k], x = qr[k], y = qi[k];
            sr += a * x + b * y;      // Re(conj(csm)*q)
            si += a * y - b * x;      // Im(conj(csm)*q)
        }
        sr += lam * pr[v];
        si += lam * pi[v];
        apr[v] = sr; api[v] = si;
        acc += pr[v] * sr + pi[v] * si;   // Re(p^H Ap)
    }
    float t = block_reduce_add(acc);
    if (threadIdx.x == 0) atomicAdd(&s[1], t);        // pAp
}

__global__ void scalar_zero(float* __restrict__ s, int idx) {
    if (threadIdx.x == 0) s[idx] = 0.0f;
}

// alpha = active ? rTr/pAp : 0 ; also clears rTrNew accumulator.
__global__ void scalar_alpha(float* __restrict__ s) {
    if (threadIdx.x == 0) {
        float rTr = s[0], pAp = s[1];
        bool act = rTr > CG_TOL;
        float d = act ? pAp : 1.0f;
        s[3] = act ? (rTr / d) : 0.0f;
        s[2] = 0.0f;
    }
}

// x += a*p ; r -= a*Ap ; fused rn·rn reduction.
__global__ void axpy_reduce(float* __restrict__ xr, float* __restrict__ xi,
                            float* __restrict__ rr, float* __restrict__ ri,
                            const float* __restrict__ pr, const float* __restrict__ pi,
                            const float* __restrict__ apr, const float* __restrict__ api,
                            float* __restrict__ s) {
    const float a = s[3];
    float acc = 0.0f;
    for (int v = blockIdx.x * blockDim.x + threadIdx.x; v < NV;
         v += gridDim.x * blockDim.x) {
        xr[v] += a * pr[v];
        xi[v] += a * pi[v];
        float nr = rr[v] - a * apr[v];
        float ni = ri[v] - a * api[v];
        rr[v] = nr; ri[v] = ni;
        acc += nr * nr + ni * ni;
    }
    float t = block_reduce_add(acc);
    if (threadIdx.x == 0) atomicAdd(&s[2], t);        // rTrNew
}

// beta = rTrNew/rTr ; gate updates; commit rTr if active.
__global__ void scalar_beta(float* __restrict__ s) {
    if (threadIdx.x == 0) {
        float rTr = s[0], rn = s[2];
        bool act = rTr > CG_TOL;
        s[4] = rn / rTr;
        s[5] = act ? 1.0f : 0.0f;
        if (act) s[0] = rn;
    }
}

__global__ void p_update(float* __restrict__ pr, float* __restrict__ pi,
                         const float* __restrict__ rr, const float* __restrict__ ri,
                         const float* __restrict__ s) {
    if (s[5] == 0.0f) return;                          // frozen (early stop)
    const float b = s[4];
    for (int v = blockIdx.x * blockDim.x + threadIdx.x; v < NV;
         v += gridDim.x * blockDim.x) {
        pr[v] = rr[v] + b * pr[v];
        pi[v] = ri[v] + b * pi[v];
    }
}

__global__ void writeback(const float* __restrict__ xr, const float* __restrict__ xi,
                          float* __restrict__ out) {
    for (int v = blockIdx.x * blockDim.x + threadIdx.x; v < NV;
         v += gridDim.x * blockDim.x) {
        out[2 * v]     = xr[v];
        out[2 * v + 1] = xi[v];
    }
}

// ------------------------------- launcher -----------------------------------
extern "C" void kernel_launch(void* const* d_in, const int* in_sizes, int n_in,
                              void* d_out, int out_size, void* d_ws, size_t ws_size,
                              hipStream_t stream) {
    (void)in_sizes; (void)n_in; (void)out_size; (void)ws_size;
    const float* rhs   = (const float*)d_in[0];   // [B,X,Y,Z,1,2]
    const float* csm_r = (const float*)d_in[1];   // [B,C,X,Y,Z]
    const float* csm_i = (const float*)d_in[2];
    const float* mask  = (const float*)d_in[3];   // [B,X,Y,Z,1]
    const float* lam1  = (const float*)d_in[4];
    const float* lam2  = (const float*)d_in[5];
    float* out = (float*)d_out;

    // ---- workspace layout (floats) ----
    float* ws = (float*)d_ws;
    float* W128fr = ws;                 ws += 128 * 128;   // forward DFT tables
    float* W128fi = ws;                 ws += 128 * 128;
    float* W128ir = ws;                 ws += 128 * 128;   // inverse
    float* W128ii = ws;                 ws += 128 * 128;
    float* W64fr  = ws;                 ws += 64 * 64;
    float* W64fi  = ws;                 ws += 64 * 64;
    float* W64ir  = ws;                 ws += 64 * 64;
    float* W64ii  = ws;                 ws += 64 * 64;
    float* scal   = ws;                 ws += 64;          // [rTr,pAp,rTrNew,alpha,beta,upd]
    float* xr = ws; ws += NV;  float* xi = ws; ws += NV;
    float* rr = ws; ws += NV;  float* ri = ws; ws += NV;
    float* pr = ws; ws += NV;  float* pi = ws; ws += NV;
    float* apr = ws; ws += NV; float* api = ws; ws += NV;
    float* qr = ws; ws += (long)NCOIL * NV;                // coil buffers (L2-resident)
    float* qi = ws; ws += (long)NCOIL * NV;

    const dim3 B256(256);
    const dim3 Gpt(1024);                                  // NV pointwise
    const dim3 Gc(4096);                                   // C*NV pointwise
    const dim3 GfX(512 * NCOIL), GfY(512 * NCOIL), GfZ(1024 * NCOIL);

    twiddle_init<<<64, B256, 0, stream>>>(W128fr, W128fi, 128, -1.0f);
    twiddle_init<<<64, B256, 0, stream>>>(W128ir, W128ii, 128,  1.0f);
    twiddle_init<<<16, B256, 0, stream>>>(W64fr,  W64fi,  64,  -1.0f);
    twiddle_init<<<16, B256, 0, stream>>>(W64ir,  W64ii,  64,   1.0f);

    for (int b = 0; b < NB; b++) {
        const float* rhs_b  = rhs   + (long)b * NV * 2;
        const float* cr_b   = csm_r + (long)b * NCOIL * NV;
        const float* ci_b   = csm_i + (long)b * NCOIL * NV;
        const float* mask_b = mask  + (long)b * NV;
        float* out_b = out + (long)b * NV * 2;

        scalar_zero<<<1, 32, 0, stream>>>(scal, 0);
        init_vecs<<<Gpt, B256, 0, stream>>>(rhs_b, xr, xi, rr, ri, pr, pi, scal);

        for (int it = 0; it < CG_ITERS; it++) {
            scalar_zero<<<1, 32, 0, stream>>>(scal, 1);           // pAp = 0
            coil_expand<<<Gc, B256, 0, stream>>>(cr_b, ci_b, pr, pi, qr, qi);

            // forward 3-D FFT (DFT matmul per axis, WMMA f32)
            dft_axis_kernel<<<GfX, B256, 0, stream>>>(qr, qi, W128fr, W128fi,
                128, 8192, 1, 0x7FFFFFFF, 31, 0, 512, (long)NV);
            dft_axis_kernel<<<GfY, B256, 0, stream>>>(qr, qi, W128fr, W128fi,
                128, 64, 1, 63, 6, 8192, 512, (long)NV);
            dft_axis_kernel<<<GfZ, B256, 0, stream>>>(qr, qi, W64fr, W64fi,
                64, 1, 64, 0x7FFFFFFF, 31, 0, 1024, (long)NV);

            mask_scale<<<Gc, B256, 0, stream>>>(qr, qi, mask_b);

            // inverse 3-D FFT (conjugate tables; 1/N folded into mask_scale)
            dft_axis_kernel<<<GfX, B256, 0, stream>>>(qr, qi, W128ir, W128ii,
                128, 8192, 1, 0x7FFFFFFF, 31, 0, 512, (long)NV);
            dft_axis_kernel<<<GfY, B256, 0, stream>>>(qr, qi, W128ir, W128ii,
                128, 64, 1, 63, 6, 8192, 512, (long)NV);
            dft_axis_kernel<<<GfZ, B256, 0, stream>>>(qr, qi, W64ir, W64ii,
                64, 1, 64, 0x7FFFFFFF, 31, 0, 1024, (long)NV);

            coil_combine<<<Gpt, B256, 0, stream>>>(cr_b, ci_b, qr, qi, pr, pi,
                                                   apr, api, lam1, lam2, scal);
            scalar_alpha<<<1, 32, 0, stream>>>(scal);
            axpy_reduce<<<Gpt, B256, 0, stream>>>(xr, xi, rr, ri, pr, pi,
                                                  apr, api, scal);
            scalar_beta<<<1, 32, 0, stream>>>(scal);
            p_update<<<Gpt, B256, 0, stream>>>(pr, pi, rr, ri, scal);
        }

        writeback<<<Gpt, B256, 0, stream>>>(xr, xi, out_b);
    }
}